// SparseCodingLoss_12747462934999
// MI455X (gfx1250) — compile-verified
//
#include <hip/hip_runtime.h>
#include <hip/hip_bf16.h>
#include <math.h>

typedef __attribute__((ext_vector_type(2))) float v2f;
typedef __attribute__((ext_vector_type(8))) float v8f;
typedef __attribute__((ext_vector_type(4))) unsigned int u32x4;
typedef __attribute__((ext_vector_type(4))) int i32x4;
typedef __attribute__((ext_vector_type(8))) int i32x8;

#define NSIG   8
#define NPOS   32768
#define NATOMS 256
#define SATOM  512
#define NSTEPS 16
#define PTILES (NPOS / 16)   // 2048 position tiles
#define WIN    (16 + SATOM)  // 528-float residual window per signal

// ---------------------------------------------------------------------------
// Dictionary re-normalization: dn[a] = d[a] / (||d[a]|| + 1e-8)
// ---------------------------------------------------------------------------
__launch_bounds__(512)
__global__ void norm_d_kernel(const float* __restrict__ d, float* __restrict__ dn) {
    __shared__ float red[512];
    const int a   = blockIdx.x;
    const int tid = threadIdx.x;
    float x = d[a * SATOM + tid];
    red[tid] = x * x;
    __syncthreads();
    for (int off = 256; off > 0; off >>= 1) {
        if (tid < off) red[tid] += red[tid + off];
        __syncthreads();
    }
    float inv = 1.0f / (sqrtf(red[0]) + 1e-8f);
    dn[a * SATOM + tid] = x * inv;
}

// ---------------------------------------------------------------------------
// Residual init: signals 0..3 = recon batches, 4..7 = target batches
// ---------------------------------------------------------------------------
__launch_bounds__(256)
__global__ void init_res_kernel(const float* __restrict__ recon,
                                const float* __restrict__ target,
                                float* __restrict__ res) {
    const int sig = blockIdx.y;
    const int i   = blockIdx.x * 256 + threadIdx.x;
    const float* src = (sig < 4) ? (recon + sig * NPOS) : (target + (sig - 4) * NPOS);
    res[sig * NPOS + i] = src[i];
}

// ---------------------------------------------------------------------------
// Fused 8-signal correlation (fp32 WMMA) + per-signal block argmax.
// Block = 512 threads = 16 waves; wave w owns atoms [16w,16w+16).
// Each wave keeps 8 accumulators (one per signal): each dictionary A
// fragment feeds 8 WMMAs (8x less L2 dictionary traffic).
// Inner loop is software-pipelined with ping-pong operand sets so each
// 8-WMMA group hides the other set's LDS/global load latency.
// Residual windows staged via one 2D Tensor-Data-Mover descriptor.
// ---------------------------------------------------------------------------
__launch_bounds__(512)
__global__ void corr_argmax_kernel(const float* __restrict__ res,
                                   const float* __restrict__ dn,
                                   float* __restrict__ bbv,
                                   int*   __restrict__ bbi) {
    __shared__ float lres[NSIG][WIN];   // 16.5 KB residual windows
    __shared__ float rv[NSIG][512];     // 16 KB argmax values
    __shared__ int   ri[NSIG][512];     // 16 KB argmax indices

    const int tid = threadIdx.x;
    const int p0  = blockIdx.x * 16;

#if __has_builtin(__builtin_amdgcn_tensor_load_to_lds)
    // ---- TDM staging: one 2D tile, 8 rows (signals) x 528 elems ----
    if (tid < 32) {   // single wave issues the DMA (EXEC ignored by TDM)
        unsigned long long gaddr = (unsigned long long)(const void*)(res + p0);
        unsigned int ldsa = (unsigned int)(unsigned long long)(const void*)&lres[0][0];

        u32x4 g0;
        g0[0] = 1u;                                            // count=1, user desc
        g0[1] = ldsa;                                          // lds_addr (bytes)
        g0[2] = (unsigned int)(gaddr & 0xffffffffull);         // global_addr lo
        g0[3] = (unsigned int)((gaddr >> 32) & 0x1ffffffull)   // global_addr hi
              | (2u << 30);                                    // type = 2 (image)

        const int tdim0 = NPOS - p0;                           // x-OOB -> zero fill
        i32x8 g1;
        g1[0] = 0x00020000;                                    // data_size = 4B
        g1[1] = (tdim0 & 0xffff) << 16;                        // tensor_dim0 lo16
        g1[2] = ((tdim0 >> 16) & 0xffff) | (NSIG << 16);       // dim0 hi | tensor_dim1=8
        g1[3] = (WIN << 16);                                   // tile_dim0 = 528
        g1[4] = NSIG;                                          // tile_dim1 = 8
        g1[5] = NPOS;                                          // tensor_dim0_stride lo
        g1[6] = 0;
        g1[7] = 0;

        i32x4 gz = {0, 0, 0, 0};
#if defined(__clang_major__) && (__clang_major__ >= 23)
        i32x8 gz8 = {0, 0, 0, 0, 0, 0, 0, 0};
        __builtin_amdgcn_tensor_load_to_lds(g0, g1, gz, gz, gz8, 0);
#else
        __builtin_amdgcn_tensor_load_to_lds(g0, g1, gz, gz, 0);
#endif
        __builtin_amdgcn_s_wait_tensorcnt(0);
    }
    __syncthreads();
#else
    // ---- Fallback: cooperative loads ----
    for (int i = tid; i < NSIG * WIN; i += 512) {
        int g = i / WIN, x = i - g * WIN;
        int p = p0 + x;
        lres[g][x] = (p < NPOS) ? res[g * NPOS + p] : 0.0f;
    }
    __syncthreads();
#endif

    const int lane  = tid & 31;
    const int wave  = tid >> 5;
    const int aBase = wave * 16;     // 16 waves cover 256 atoms
    const int m     = lane & 15;     // A-row / B-col index for this lane
    const int h     = lane >> 4;     // K-pair half select

    const float* drow = dn + (aBase + m) * SATOM;
    const int bo = m + 2 * h;        // B base float offset within a window row

    v8f acc[NSIG];
    #pragma unroll
    for (int g = 0; g < NSIG; ++g) acc[g] = (v8f){};

    // ---- software-pipelined ping-pong over 128 K-chunks of 4 ----
    v2f aA, aB;
    v2f bA[NSIG], bB[NSIG];

    aA = *(const v2f*)(drow + 2 * h);                 // chunk 0
    #pragma unroll
    for (int g = 0; g < NSIG; ++g) {
        bA[g].x = lres[g][bo];
        bA[g].y = lres[g][bo + 1];
    }

    for (int c = 0; c < 128; c += 2) {
        const int s1 = 4 * (c + 1);
        const int s2 = 4 * (c + 2);   // c=126 -> chunk 128: harmless over-read,
                                      // values never consumed
        // load chunk c+1 into set B
        aB = *(const v2f*)(drow + s1 + 2 * h);
        #pragma unroll
        for (int g = 0; g < NSIG; ++g) {
            bB[g].x = lres[g][bo + s1];
            bB[g].y = lres[g][bo + s1 + 1];
        }
        // 8 WMMAs on set A (chunk c) — hides set-B load latency
        #pragma unroll
        for (int g = 0; g < NSIG; ++g)
            acc[g] = __builtin_amdgcn_wmma_f32_16x16x4_f32(
                false, aA, false, bA[g], (short)0, acc[g], false, false);

        // load chunk c+2 into set A
        aA = *(const v2f*)(drow + s2 + 2 * h);
        #pragma unroll
        for (int g = 0; g < NSIG; ++g) {
            bA[g].x = lres[g][bo + s2];
            bA[g].y = lres[g][bo + s2 + 1];
        }
        // 8 WMMAs on set B (chunk c+1) — hides set-A load latency
        #pragma unroll
        for (int g = 0; g < NSIG; ++g)
            acc[g] = __builtin_amdgcn_wmma_f32_16x16x4_f32(
                false, aB, false, bB[g], (short)0, acc[g], false, false);
    }

    // Per-lane, per-signal argmax over the 8 accumulator rows; first-max
    // tie-break (lower row == lower atom == lower flat index, p fixed).
    #pragma unroll
    for (int g = 0; g < NSIG; ++g) {
        float bv = acc[g][0];
        int   br = 0;
        #pragma unroll
        for (int rr = 1; rr < 8; ++rr) {
            if (acc[g][rr] > bv) { bv = acc[g][rr]; br = rr; }
        }
        rv[g][tid] = bv;
        ri[g][tid] = (aBase + br + 8 * h) * NPOS + (p0 + m);
    }
    __syncthreads();

    // 8 parallel tree reductions (tie-break: lowest flat index)
    for (int off = 256; off > 0; off >>= 1) {
        if (tid < off) {
            #pragma unroll
            for (int g = 0; g < NSIG; ++g) {
                float v2 = rv[g][tid + off];
                int   i2 = ri[g][tid + off];
                if (v2 > rv[g][tid] || (v2 == rv[g][tid] && i2 < ri[g][tid])) {
                    rv[g][tid] = v2; ri[g][tid] = i2;
                }
            }
        }
        __syncthreads();
    }
    if (tid < NSIG) {
        bbv[tid * PTILES + blockIdx.x] = rv[tid][0];
        bbi[tid * PTILES + blockIdx.x] = ri[tid][0];
    }
}

// ---------------------------------------------------------------------------
// Final argmax over block-bests, record pick, subtract v*d[a] from residual.
// One block per signal.
// ---------------------------------------------------------------------------
__launch_bounds__(512)
__global__ void pick_update_kernel(const float* __restrict__ bbv,
                                   const int*   __restrict__ bbi,
                                   float* __restrict__ res,
                                   const float* __restrict__ dn,
                                   int* __restrict__ pa, int* __restrict__ pp,
                                   float* __restrict__ pv, int step) {
    __shared__ float rv[512];
    __shared__ int   ri[512];
    __shared__ int   s_a, s_p;
    __shared__ float s_v;

    const int sig = blockIdx.x;
    const int tid = threadIdx.x;

    float bv = -INFINITY;
    int   bi = 0x7fffffff;
    for (int j = tid; j < PTILES; j += 512) {
        float v = bbv[sig * PTILES + j];
        int   i = bbi[sig * PTILES + j];
        if (v > bv || (v == bv && i < bi)) { bv = v; bi = i; }
    }
    rv[tid] = bv; ri[tid] = bi;
    __syncthreads();
    for (int off = 256; off > 0; off >>= 1) {
        if (tid < off) {
            float v2 = rv[tid + off];
            int   i2 = ri[tid + off];
            if (v2 > rv[tid] || (v2 == rv[tid] && i2 < ri[tid])) {
                rv[tid] = v2; ri[tid] = i2;
            }
        }
        __syncthreads();
    }
    if (tid == 0) {
        int   idx = ri[0];
        float v   = rv[0];
        int a = idx / NPOS;
        int p = idx - a * NPOS;
        s_a = a; s_p = p; s_v = v;
        pa[sig * NSTEPS + step] = a;
        pp[sig * NSTEPS + step] = p;
        pv[sig * NSTEPS + step] = v;
    }
    __syncthreads();

    const int a = s_a, p = s_p;
    const float v = s_v;
    const int s = tid;                 // 512 threads, one tap each
    if (p + s < NPOS) {
        res[sig * NPOS + p + s] -= v * dn[a * SATOM + s];
    }
}

// ---------------------------------------------------------------------------
// Sparse BCE: accumulate duplicate picks, joint max, per-cell BCE terms in
// float (mirrors reference f32 clip semantics), background term counted
// analytically, double accumulator.
// ---------------------------------------------------------------------------
__global__ void loss_kernel(const int* __restrict__ pa, const int* __restrict__ pp,
                            const float* __restrict__ pv, float* __restrict__ out) {
    if (threadIdx.x != 0 || blockIdx.x != 0) return;

    int   keys[NSIG][NSTEPS];
    float vals[NSIG][NSTEPS];
    int   cnt[NSIG];

    for (int s = 0; s < NSIG; ++s) {
        cnt[s] = 0;
        for (int k = 0; k < NSTEPS; ++k) {
            int   key = pa[s * NSTEPS + k] * NPOS + pp[s * NSTEPS + k];
            float v   = pv[s * NSTEPS + k];
            int j = 0;
            for (; j < cnt[s]; ++j) {
                if (keys[s][j] == key) { vals[s][j] += v; break; }
            }
            if (j == cnt[s]) { keys[s][j] = key; vals[s][j] = v; cnt[s]++; }
        }
    }

    float mx = 0.0f;   // zero background participates in the max
    for (int s = 0; s < NSIG; ++s)
        for (int j = 0; j < cnt[s]; ++j)
            if (vals[s][j] > mx) mx = vals[s][j];

    double total = 0.0;
    const double bg_term = -(double)log1pf(-1e-12f);  // per all-zero cell

    for (int b = 0; b < 4; ++b) {
        const int rs = b, ts = b + 4;
        long nU = 0;
        for (int j = 0; j < cnt[rs]; ++j) {
            float rr = vals[rs][j] / mx;
            float tt = 0.0f;
            for (int i = 0; i < cnt[ts]; ++i)
                if (keys[ts][i] == keys[rs][j]) { tt = vals[ts][i] / mx; break; }
            float rc = fminf(fmaxf(rr, 1e-12f), 1.0f - 1e-12f);
            float lr = fmaxf(logf(rc), -100.0f);
            float l1 = fmaxf(log1pf(-rc), -100.0f);
            total += -(double)(tt * lr + (1.0f - tt) * l1);
            nU++;
        }
        for (int i = 0; i < cnt[ts]; ++i) {
            bool inR = false;
            for (int j = 0; j < cnt[rs]; ++j)
                if (keys[rs][j] == keys[ts][i]) { inR = true; break; }
            if (!inR) {
                float tt = vals[ts][i] / mx;
                float rc = 1e-12f;
                float lr = fmaxf(logf(rc), -100.0f);
                float l1 = fmaxf(log1pf(-rc), -100.0f);
                total += -(double)(tt * lr + (1.0f - tt) * l1);
                nU++;
            }
        }
        total += ((double)NATOMS * (double)NPOS - (double)nU) * bg_term;
    }

    out[0] = (float)(total / ((double)4 * (double)NATOMS * (double)NPOS));
}

// ---------------------------------------------------------------------------
extern "C" void kernel_launch(void* const* d_in, const int* in_sizes, int n_in,
                              void* d_out, int out_size, void* d_ws, size_t ws_size,
                              hipStream_t stream) {
    const float* recon  = (const float*)d_in[0];   // (4,1,32768) f32
    const float* target = (const float*)d_in[1];   // (4,1,32768) f32
    const float* d      = (const float*)d_in[2];   // (256,512)  f32

    char* ws = (char*)d_ws;
    float* dn  = (float*)ws;  ws += (size_t)NATOMS * SATOM * sizeof(float);   // 512 KB
    float* res = (float*)ws;  ws += (size_t)NSIG * NPOS * sizeof(float);      // 1 MB
    float* bbv = (float*)ws;  ws += (size_t)NSIG * PTILES * sizeof(float);    // 64 KB
    int*   bbi = (int*)ws;    ws += (size_t)NSIG * PTILES * sizeof(int);      // 64 KB
    int*   pa  = (int*)ws;    ws += (size_t)NSIG * NSTEPS * sizeof(int);
    int*   pp  = (int*)ws;    ws += (size_t)NSIG * NSTEPS * sizeof(int);
    float* pv  = (float*)ws;  ws += (size_t)NSIG * NSTEPS * sizeof(float);

    norm_d_kernel<<<NATOMS, 512, 0, stream>>>(d, dn);
    init_res_kernel<<<dim3(NPOS / 256, NSIG), 256, 0, stream>>>(recon, target, res);

    for (int step = 0; step < NSTEPS; ++step) {
        corr_argmax_kernel<<<dim3(PTILES), 512, 0, stream>>>(res, dn, bbv, bbi);
        pick_update_kernel<<<NSIG, 512, 0, stream>>>(bbv, bbi, res, dn, pa, pp, pv, step);
    }

    loss_kernel<<<1, 32, 0, stream>>>(pa, pp, pv, (float*)d_out);
}